// Lift_73650099192096
// MI455X (gfx1250) — compile-verified
//
#include <hip/hip_runtime.h>

// Lift: out[r, 0:1024] = x[r, 0:1024]; out[r, 1024:4096] = 0
// x: (32768, 1024) f32, out: (32768, 4096) f32.
// Pure bandwidth problem: 128 MiB in + 512 MiB out = 640 MiB => ~28 us @ 23.3 TB/s.
// Copy region goes through the CDNA5 async global<->LDS mover (ASYNCcnt path);
// pad region is non-temporal B128 zero stores. All traffic is single-touch => NT everywhere.

typedef __attribute__((ext_vector_type(4))) float vfloat4;
typedef int v4i __attribute__((vector_size(16)));               // int4, matches builtin pointee
typedef __attribute__((address_space(1))) v4i gv4i;             // global int4
typedef __attribute__((address_space(3))) v4i lv4i;             // LDS int4

#define D_IN   1024
#define D_OUT  4096
#define TPB    256
#define CPOL_TH_NT 1   // temporal-hint NT in cpol bits [2:0]

#if defined(__AMDGCN__) && \
    __has_builtin(__builtin_amdgcn_global_load_async_to_lds_b128) && \
    __has_builtin(__builtin_amdgcn_global_store_async_from_lds_b128) && \
    __has_builtin(__builtin_amdgcn_s_wait_asynccnt)
#define LIFT_USE_ASYNC 1
#else
#define LIFT_USE_ASYNC 0
#endif

__global__ __launch_bounds__(TPB) void lift_pad_kernel(const float* __restrict__ x,
                                                       float* __restrict__ out)
{
    const unsigned row = blockIdx.x;   // one row per block
    const unsigned t   = threadIdx.x;  // 0..255

    const vfloat4* __restrict__ in4 = (const vfloat4*)(x + (size_t)row * D_IN);
    vfloat4* __restrict__ out4      = (vfloat4*)(out + (size_t)row * D_OUT);

    // ---- copy region: cols [0, 1024) == 256 float4s, one per thread ----
#if LIFT_USE_ASYNC
    __shared__ vfloat4 stage[TPB];  // 4 KB staging; each wave owns a private 512 B slice
    gv4i* gsrc = (gv4i*)(in4 + t);
    gv4i* gdst = (gv4i*)(out4 + t);
    lv4i* lptr = (lv4i*)(&stage[t]);
    // HBM -> LDS, async (no VGPR round-trip); tracked by ASYNCcnt; NT (read-once)
    __builtin_amdgcn_global_load_async_to_lds_b128(gsrc, lptr, 0, CPOL_TH_NT);
    // Each lane's store reads only LDS its own wave's load wrote -> per-wave wait, no barrier
    __builtin_amdgcn_s_wait_asynccnt(0);
    // LDS -> HBM, async; NT (write-once streaming)
    __builtin_amdgcn_global_store_async_from_lds_b128(gdst, lptr, 0, CPOL_TH_NT);
#else
    vfloat4 v = __builtin_nontemporal_load(in4 + t);
    __builtin_nontemporal_store(v, out4 + t);
#endif

    // ---- pad region: cols [1024, 4096) == 768 float4s, 3 per thread ----
    // Streaming zeros: non-temporal so 384 MiB of write-once data skips cache residency.
    const vfloat4 z = {0.0f, 0.0f, 0.0f, 0.0f};
    __builtin_nontemporal_store(z, out4 + 1 * TPB + t);
    __builtin_nontemporal_store(z, out4 + 2 * TPB + t);
    __builtin_nontemporal_store(z, out4 + 3 * TPB + t);
}

extern "C" void kernel_launch(void* const* d_in, const int* in_sizes, int n_in,
                              void* d_out, int out_size, void* d_ws, size_t ws_size,
                              hipStream_t stream)
{
    const float* x = (const float*)d_in[0];
    float* out = (float*)d_out;
    const int rows = in_sizes[0] / D_IN;  // 32768
    lift_pad_kernel<<<dim3(rows), dim3(TPB), 0, stream>>>(x, out);
}